// KPConv_46136538694256
// MI455X (gfx1250) — compile-verified
//
#include <hip/hip_runtime.h>

// ---------------------------------------------------------------------------
// KPConv fused kernel for MI455X (gfx1250), fp32 end-to-end via
// V_WMMA_F32_16X16X4_F32.
//
// Shapes: B=4, N=16384, S=16 neighbors, K=15 kernel points (padded to 16),
//         C=64 in channels, O=64 out channels.
// ---------------------------------------------------------------------------

typedef __attribute__((ext_vector_type(2))) float v2f;
typedef __attribute__((ext_vector_type(8))) float v8f;

#define BATCH   4
#define NPTS    16384
#define NSLOT   16
#define NKP     15
#define CH      64
#define OCH     64
#define KP_EXT  0.06f
#define BIGV    1000000.0f

// d_out layout (flat float elements), tuple (out, p, x_in_skip, n_idx):
#define OUT_MAIN 0
#define OUT_P    4194304      // B*O*N
#define OUT_SKIP 4390912      // + B*N*3
#define OUT_NIDX 8585216      // + B*C*N

// ---------------------------------------------------------------------------
// Kernel 0: (B,C,N) -> (B,N,C) transpose so neighbor gathers become 256B
// contiguous row reads instead of stride-64KB column gathers.
// ---------------------------------------------------------------------------
__global__ __launch_bounds__(256)
void kp_transpose(const float* __restrict__ src, float* __restrict__ dst) {
    __shared__ float tile[64][65];           // +1 pad: no bank conflicts
    const int b  = blockIdx.x >> 8;          // 256 n-tiles of 64 per batch
    const int n0 = (blockIdx.x & 255) << 6;
    const float* s = src + (size_t)b * CH * NPTS;
    float*       d = dst + (size_t)b * CH * NPTS;

    for (int e = threadIdx.x; e < 4096; e += 256) {
        int c = e >> 6, j = e & 63;          // coalesced along n
        tile[c][j] = s[(size_t)c * NPTS + n0 + j];
    }
    __syncthreads();
    for (int e = threadIdx.x; e < 4096; e += 256) {
        int j = e >> 6, c = e & 63;          // coalesced along c
        d[(size_t)(n0 + j) * CH + c] = tile[c][j];
    }
}

// ---------------------------------------------------------------------------
// Main fused kernel: one workgroup (128 threads = 4 waves) per 16-point tile.
// ---------------------------------------------------------------------------
__global__ __launch_bounds__(128)
void kpconv_main(const float* __restrict__ p,
                 const int*   __restrict__ n_idx,
                 const float* __restrict__ kp,
                 const float* __restrict__ wts,    // (15,64,64) flat = (kc,o)
                 const float* __restrict__ xT,     // (B,N,C)
                 const float* __restrict__ xinT,   // (B,N,C)
                 float* __restrict__ out,          // (B,O,N)
                 float* __restrict__ skipOut)      // (B,C,N)
{
    __shared__ int    s_idx[16][16];
    __shared__ float  s_pc[16][4];
    __shared__ float4 s_rel[16][16];   // xyz + bitcast byte-offset into xT
    __shared__ float  s_wt[16][1024];  // weighted[point][k*64+c], 64 KB
    __shared__ float  s_out[64][16];   // out tile [o][i] for coalesced store

    const int tid = threadIdx.x;
    const int b   = blockIdx.x >> 10;          // 1024 tiles of 16 per batch
    const int n0  = (blockIdx.x & 1023) << 4;

    // L2-warm the (small, shared) weight matrix.
    __builtin_prefetch(wts + tid * 64, 0, 1);

    // ---- Phase 0: neighbor indices + center points --------------------------
    for (int e = tid; e < 256; e += 128) {
        int i = e >> 4, s = e & 15;
        s_idx[i][s] = n_idx[((b * NPTS + n0 + i) << 4) + s];
    }
    if (tid < 48) {
        int i = tid / 3, d = tid - i * 3;
        s_pc[i][d] = p[(b * NPTS + n0 + i) * 3 + d];
    }
    __syncthreads();

    // ---- Phase 1: relative xyz (BIG for padded slots) + feature offsets ----
    for (int e = tid; e < 256; e += 128) {
        int i = e >> 4, s = e & 15;
        int idx = s_idx[i][s];
        bool pad = (s > 0) && (idx == s_idx[i][0]);
        const float* gp = p + (b * NPTS + idx) * 3;
        float4 r;
        r.x = pad ? BIGV : (gp[0] - s_pc[i][0]);
        r.y = pad ? BIGV : (gp[1] - s_pc[i][1]);
        r.z = pad ? BIGV : (gp[2] - s_pc[i][2]);
        r.w = __uint_as_float((unsigned)(b * NPTS + idx) * (CH * 4u));
        s_rel[i][s] = r;
    }
    __syncthreads();

    // ---- Phase 2: skip connection: max over masked neighbors ---------------
    {
        int i  = tid >> 3;           // point
        int cb = (tid & 7) << 3;     // 8 channels per thread
        float m[8];
        #pragma unroll
        for (int r = 0; r < 8; ++r) m[r] = -3.402823466e38f;
        #pragma unroll
        for (int s = 0; s < NSLOT; ++s) {
            float4 r = s_rel[i][s];
            bool pad = r.x >= 1.0e5f;              // BIG sentinel
            const float4* row =
                (const float4*)(xinT + (__float_as_uint(r.w) >> 2) + cb);
            float4 v0 = row[0], v1 = row[1];
            float v[8] = {v0.x, v0.y, v0.z, v0.w, v1.x, v1.y, v1.z, v1.w};
            #pragma unroll
            for (int rr = 0; rr < 8; ++rr)
                m[rr] = fmaxf(m[rr], pad ? 0.0f : v[rr]);
        }
        #pragma unroll
        for (int rr = 0; rr < 8; ++rr)
            skipOut[(size_t)(b * CH + cb + rr) * NPTS + n0 + i] = m[rr];
    }

    // ---- Stage A: weighted[i,k,c] = sum_s all_w[i,s,k] * feat[i,c,s] -------
    // WMMA mapping per point: M=k(16, row 15 zeroed), K=s(16), N=c(64).
    const int lane = tid & 31;
    const int wv   = tid >> 5;
    const int half = lane >> 4;      // fp32 A/B: K split across lane halves
    const int cl   = lane & 15;

    // kernel point for this lane's matrix row (branch-free, k=15 -> weight 0)
    const int  kidx   = (cl < NKP) ? cl : 0;
    const bool kvalid = (cl < NKP);
    const float kx = kp[kidx * 3 + 0];
    const float ky = kp[kidx * 3 + 1];
    const float kz = kp[kidx * 3 + 2];

    for (int pt = 0; pt < 4; ++pt) {
        const int i = (wv << 2) + pt;
        float    aw[16];
        unsigned off[16];
        #pragma unroll
        for (int s = 0; s < 16; ++s) {
            float4 r = s_rel[i][s];
            off[s] = __float_as_uint(r.w);
            float dx = r.x - kx, dy = r.y - ky, dz = r.z - kz;
            float d  = sqrtf(dx * dx + dy * dy + dz * dz);
            float w  = fmaxf(1.0f - d * (1.0f / KP_EXT), 0.0f);
            aw[s] = kvalid ? w : 0.0f;    // padded slots: d huge -> w = 0
        }
        // A fragments (16x4 fp32 layout: VGPR0 = K {s0, s0+2}, VGPR1 = +1/+3)
        v2f      afr[4];
        unsigned oa[4], ob[4];
        #pragma unroll
        for (int q = 0; q < 4; ++q) {
            int s0 = q * 4;
            afr[q].x = half ? aw[s0 + 2] : aw[s0 + 0];
            afr[q].y = half ? aw[s0 + 3] : aw[s0 + 1];
            oa[q]    = half ? off[s0 + 2] : off[s0 + 0];
            ob[q]    = half ? off[s0 + 3] : off[s0 + 1];
        }
        #pragma unroll
        for (int t = 0; t < 4; ++t) {            // 4 channel tiles of 16
            const int c = (t << 4) + cl;
            v8f acc = {};
            #pragma unroll
            for (int q = 0; q < 4; ++q) {        // K=16 in chunks of 4
                v2f bfr;                          // B row = slot s, col = c
                bfr.x = xT[(oa[q] >> 2) + c];
                bfr.y = xT[(ob[q] >> 2) + c];
                acc = __builtin_amdgcn_wmma_f32_16x16x4_f32(
                        false, afr[q], false, bfr, (short)0, acc, false, false);
            }
            #pragma unroll
            for (int j = 0; j < 8; ++j)           // D: VGPR j = rows j, j+8
                s_wt[i][(j + half * 8) * 64 + c] = acc[j];
        }
    }
    __syncthreads();

    // ---- Stage B: out[i,o] = sum_{kc<960} weighted[i,kc] * W[kc,o] ---------
    // WMMA mapping: M=i(16 points), K=kc(960), N=o. Wave wv owns o-tile wv.
    {
        const int ob_ = wv << 4;
        const float* wrow = wts + ob_ + cl;
        v8f acc = {};
        #pragma unroll 4
        for (int q = 0; q < 240; ++q) {
            const int kc = (q << 2) + (half << 1);
            v2f a = *(const v2f*)&s_wt[cl][kc];       // ds_load_b64
            v2f bf;
            bf.x = wrow[(size_t)kc << 6];             // W[kc,   o]
            bf.y = wrow[(size_t)(kc + 1) << 6];       // W[kc+1, o]
            acc = __builtin_amdgcn_wmma_f32_16x16x4_f32(
                    false, a, false, bf, (short)0, acc, false, false);
        }
        #pragma unroll
        for (int j = 0; j < 8; ++j)                   // D -> LDS [o][i]
            s_out[ob_ + cl][j + half * 8] = acc[j];
    }
    __syncthreads();

    // ---- Coalesced store of the (O=64, i=16) tile to (B,O,N) ---------------
    {
        const int o = tid >> 1, part = tid & 1;
        const float4* sp = (const float4*)&s_out[o][part * 8];
        float4 v0 = sp[0], v1 = sp[1];
        float* dst = out + (size_t)(b * OCH + o) * NPTS + n0 + part * 8;
        ((float4*)dst)[0] = v0;
        ((float4*)dst)[1] = v1;
    }
}

// ---------------------------------------------------------------------------
// Launch
// ---------------------------------------------------------------------------
extern "C" void kernel_launch(void* const* d_in, const int* in_sizes, int n_in,
                              void* d_out, int out_size, void* d_ws, size_t ws_size,
                              hipStream_t stream) {
    (void)in_sizes; (void)n_in; (void)out_size; (void)ws_size;

    const float* p     = (const float*)d_in[0];
    const float* x     = (const float*)d_in[1];
    const float* x_in  = (const float*)d_in[2];
    const int*   nidx  = (const int*)d_in[3];
    const float* kpnts = (const float*)d_in[4];
    const float* wts   = (const float*)d_in[5];
    float*       out   = (float*)d_out;

    float* xT   = (float*)d_ws;                         // 16 MB
    float* xinT = xT + (size_t)BATCH * CH * NPTS;       // +16 MB

    // Phase 0: transpose features for contiguous neighbor gathers.
    kp_transpose<<<BATCH * (NPTS / 64), 256, 0, stream>>>(x, xT);
    kp_transpose<<<BATCH * (NPTS / 64), 256, 0, stream>>>(x_in, xinT);

    // Pass-through outputs: p and n_idx (bit-exact copies).
    hipMemcpyAsync(out + OUT_P, p, (size_t)BATCH * NPTS * 3 * sizeof(float),
                   hipMemcpyDeviceToDevice, stream);
    hipMemcpyAsync(out + OUT_NIDX, nidx,
                   (size_t)BATCH * NPTS * NSLOT * sizeof(int),
                   hipMemcpyDeviceToDevice, stream);

    // Fused KPConv: 4096 tiles of 16 points, 4 waves each.
    kpconv_main<<<BATCH * (NPTS / 16), 128, 0, stream>>>(
        p, nidx, kpnts, wts, xT, xinT, out + OUT_MAIN, out + OUT_SKIP);
}